// Head_1468878815258
// MI455X (gfx1250) — compile-verified
//
#include <hip/hip_runtime.h>
#include <hip/hip_bf16.h>
#include <math.h>

// MI455X / gfx1250, wave32. Fused QKV projection + flash attention,
// all matmuls through V_WMMA_F32_16X16X4_F32 (fp32-exact tensor path).

typedef __attribute__((ext_vector_type(2))) float v2f;
typedef __attribute__((ext_vector_type(8))) float v8f;

__device__ __forceinline__ v8f wmma4(v2f a, v2f b, v8f c) {
  // 8 args: (neg_a, A, neg_b, B, c_mod, C, reuse_a, reuse_b)
  return __builtin_amdgcn_wmma_f32_16x16x4_f32(
      false, a, false, b, (short)0, c, false, false);
}

__device__ __forceinline__ v8f v8zero() {
  v8f z;
#pragma unroll
  for (int i = 0; i < 8; ++i) z[i] = 0.0f;
  return z;
}

#define TT 2048
#define CC 512
#define HH 64

// ---------------------------------------------------------------------------
// Kernel 1: Q = x@Wq, K = x@Wk, V = x@Wv.
// grid (T/128, B), block 256 (8 waves). Wave w -> 16 rows, all 64 cols, 3 mats.
// WMMA f32 16x16x4 layouts (wave32):
//   A 16x4 : lanes 0-15 M=lane,(K=0,1)=(a.x,a.y); lanes 16-31 (K=2,3)
//   B 4x16 : lanes n=lane%16, (b.x,b.y) = rows (2*half, 2*half+1)
//   C 16x16: vgpr r, lane: row = r + 8*(lane>=16), col = lane%16
// ---------------------------------------------------------------------------
__global__ __launch_bounds__(256) void qkv_proj_kernel(
    const float* __restrict__ x,
    const float* __restrict__ Wq, const float* __restrict__ Wk,
    const float* __restrict__ Wv,
    float* __restrict__ Q, float* __restrict__ K, float* __restrict__ V) {
  const int b    = blockIdx.y;
  const int lane = threadIdx.x & 31;
  const int wave = threadIdx.x >> 5;
  const int half = lane >> 4;
  const int ln   = lane & 15;
  const int i0   = blockIdx.x * 128 + wave * 16;

  const float* xrow = x + ((size_t)b * TT + (i0 + ln)) * CC;

  v8f aq[4], ak[4], av[4];
#pragma unroll
  for (int nt = 0; nt < 4; ++nt) { aq[nt] = v8zero(); ak[nt] = v8zero(); av[nt] = v8zero(); }

  for (int kk = 0; kk < CC; kk += 4) {
    v2f a;
    a.x = xrow[kk + 2 * half];
    a.y = xrow[kk + 2 * half + 1];
#pragma unroll
    for (int nt = 0; nt < 4; ++nt) {
      const int n  = nt * 16 + ln;
      const int r0 = (kk + 2 * half) * HH + n;
      v2f bq, bk, bv;
      bq.x = Wq[r0]; bq.y = Wq[r0 + HH];
      bk.x = Wk[r0]; bk.y = Wk[r0 + HH];
      bv.x = Wv[r0]; bv.y = Wv[r0 + HH];
      aq[nt] = wmma4(a, bq, aq[nt]);
      ak[nt] = wmma4(a, bk, ak[nt]);
      av[nt] = wmma4(a, bv, av[nt]);
    }
  }

#pragma unroll
  for (int nt = 0; nt < 4; ++nt) {
#pragma unroll
    for (int r = 0; r < 8; ++r) {
      const int row = i0 + r + 8 * half;
      const int col = nt * 16 + ln;
      const size_t o = ((size_t)b * TT + row) * HH + col;
      Q[o] = aq[nt][r];
      K[o] = ak[nt][r];
      V[o] = av[nt][r];
    }
  }
}

// ---------------------------------------------------------------------------
// Kernel 2: flash attention with band-sparse mask.
// grid (T/128, B), block 256 (8 waves). Wave -> 16 query rows, online softmax
// over 16-wide key tiles. Mask: j < (i/6+1)*6 || j >= T-6.
// ---------------------------------------------------------------------------
__global__ __launch_bounds__(256) void flash_attn_kernel(
    const float* __restrict__ Q, const float* __restrict__ K,
    const float* __restrict__ V, float* __restrict__ out) {
  const float scale = 0.04419417382415922f;  // C^-0.5 = 1/sqrt(512)
  __shared__ float pbuf[8][16][17];          // per-wave P tile, padded

  const int b    = blockIdx.y;
  const int lane = threadIdx.x & 31;
  const int wave = threadIdx.x >> 5;
  const int half = lane >> 4;
  const int ln   = lane & 15;
  const int i0   = blockIdx.x * 128 + wave * 16;

  // Q strip as 16 A-fragments (H = 64 = 16 * 4)
  v2f qa[16];
  const float* qrow = Q + ((size_t)b * TT + (i0 + ln)) * HH;
#pragma unroll
  for (int kh = 0; kh < 16; ++kh) {
    qa[kh].x = qrow[kh * 4 + 2 * half];
    qa[kh].y = qrow[kh * 4 + 2 * half + 1];
  }

  v8f o[4];
#pragma unroll
  for (int nt = 0; nt < 4; ++nt) o[nt] = v8zero();
  float m[8], l[8];
#pragma unroll
  for (int r = 0; r < 8; ++r) { m[r] = -1e30f; l[r] = 0.0f; }

  // band limit for this wave's 16 rows (monotone in i) + always do last tile
  const int band_end   = ((i0 + 15) / 6 + 1) * 6;
  int ntile_band       = (band_end + 15) / 16;
  if (ntile_band > TT / 16) ntile_band = TT / 16;
  const int last_tile  = TT / 16 - 1;

  for (int kt = 0; kt < TT / 16; ++kt) {
    if (kt >= ntile_band && kt != last_tile) continue;
    const int j0 = kt * 16;

    // ---- S = Q @ K^T (16x16), reduction over H ----
    v8f s = v8zero();
    const float* kbase = K + ((size_t)b * TT + (j0 + ln)) * HH;
#pragma unroll
    for (int kh = 0; kh < 16; ++kh) {
      v2f bk;
      bk.x = kbase[kh * 4 + 2 * half];
      bk.y = kbase[kh * 4 + 2 * half + 1];
      s = wmma4(qa[kh], bk, s);
    }

    // ---- mask + online softmax ----
    float alpha[8];
#pragma unroll
    for (int r = 0; r < 8; ++r) {
      const int i = i0 + r + 8 * half;
      const int j = j0 + ln;
      const bool ok = (j < (i / 6 + 1) * 6) || (j >= TT - 6);
      const float sv = ok ? s[r] * scale : -1e30f;

      float mx = sv;
      mx = fmaxf(mx, __shfl_xor(mx, 1));
      mx = fmaxf(mx, __shfl_xor(mx, 2));
      mx = fmaxf(mx, __shfl_xor(mx, 4));
      mx = fmaxf(mx, __shfl_xor(mx, 8));
      const float mnew = fmaxf(m[r], mx);

      const float pv = __expf(sv - mnew);
      float rs = pv;
      rs += __shfl_xor(rs, 1);
      rs += __shfl_xor(rs, 2);
      rs += __shfl_xor(rs, 4);
      rs += __shfl_xor(rs, 8);

      alpha[r] = __expf(m[r] - mnew);
      l[r] = l[r] * alpha[r] + rs;
      m[r] = mnew;

      pbuf[wave][r + 8 * half][ln] = pv;  // stage P (C-layout -> LDS)
    }

    // rescale running output
#pragma unroll
    for (int nt = 0; nt < 4; ++nt)
#pragma unroll
      for (int r = 0; r < 8; ++r) o[nt][r] *= alpha[r];

    // wave-private LDS RAW fence
    asm volatile("s_wait_dscnt 0x0" ::: "memory");

    // ---- O += P @ V (reduction over 16 keys) ----
    const float* vbase = V + ((size_t)b * TT + j0) * HH;
#pragma unroll
    for (int kc = 0; kc < 4; ++kc) {
      v2f pa;
      pa.x = pbuf[wave][ln][kc * 4 + 2 * half];
      pa.y = pbuf[wave][ln][kc * 4 + 2 * half + 1];
#pragma unroll
      for (int nt = 0; nt < 4; ++nt) {
        const float* vr = vbase + (size_t)(kc * 4 + 2 * half) * HH + nt * 16 + ln;
        v2f bv;
        bv.x = vr[0];
        bv.y = vr[HH];
        o[nt] = wmma4(pa, bv, o[nt]);
      }
    }
  }

  // ---- normalize + store ----
#pragma unroll
  for (int nt = 0; nt < 4; ++nt) {
#pragma unroll
    for (int r = 0; r < 8; ++r) {
      const int row = i0 + r + 8 * half;
      const int col = nt * 16 + ln;
      out[((size_t)b * TT + row) * HH + col] = o[nt][r] / l[r];
    }
  }
}

// ---------------------------------------------------------------------------
extern "C" void kernel_launch(void* const* d_in, const int* in_sizes, int n_in,
                              void* d_out, int out_size, void* d_ws, size_t ws_size,
                              hipStream_t stream) {
  const float* x  = (const float*)d_in[0];
  const float* Wq = (const float*)d_in[1];
  const float* Wk = (const float*)d_in[2];
  const float* Wv = (const float*)d_in[3];
  float* out = (float*)d_out;

  const int B = 16;
  const size_t n = (size_t)B * TT * HH;  // 2M elems per tensor
  float* Qw = (float*)d_ws;              // needs 3*n*4 = 24 MB of workspace
  float* Kw = Qw + n;
  float* Vw = Kw + n;

  dim3 grid(TT / 128, B);
  dim3 block(256);
  qkv_proj_kernel<<<grid, block, 0, stream>>>(x, Wq, Wk, Wv, Qw, Kw, Vw);
  flash_attn_kernel<<<grid, block, 0, stream>>>(Qw, Kw, Vw, out);
}